// EmbeddingCRF_36309653520531
// MI455X (gfx1250) — compile-verified
//
#include <hip/hip_runtime.h>
#include <hip/hip_bf16.h>

typedef float v2f __attribute__((ext_vector_type(2)));
typedef float v8f __attribute__((ext_vector_type(8)));

#define EMB 300
#define NTAGS 14
#define SEQL 32768
#define NEGV -10000.0f
#define STARTT 12
#define STOPT 13
#define CHUNK 128
#define NCHUNK (SEQL / CHUNK)   // 256

// ---- workspace layout (bytes) ----
#define OFF_FEATS 0                         // SEQL*16 f32   = 2,097,152
#define OFF_MG    2097152                   // NCHUNK*256 f32 = 262,144
#define OFF_PG    2359296                   // NCHUNK*256 f32 = 262,144
#define OFF_PVB   2621440                   // NCHUNK*16 f32  = 16,384
#define OFF_FMAP  2637824                   // NCHUNK*16 i32  = 16,384
#define OFF_ENTRY 2654208                   // NCHUNK i32     = 1,024
#define OFF_BP    2655232                   // SEQL u64       = 262,144 (8B aligned)
#define OFF_GOLD  2917376                   // 64 f32
#define OFF_SCALI 2917632                   // ints

// ============================================================
// K1: feats[t][n] = sum_k emb[sentence[t]][k] * W[n][k]
// One wave per 16-row tile, V_WMMA_F32_16X16X4_F32, K-loop of 75.
// W is staged (transposed, zero-padded to 16 tags) into LDS as
// (k/2, n) float2 pairs so each B fragment is one ds_load_b64.
// ============================================================
__global__ __launch_bounds__(256)
void feats_wmma(const int* __restrict__ sentence,
                const float* __restrict__ emb,
                const float* __restrict__ W,
                float* __restrict__ feats) {
  __shared__ v2f wt[150 * 16];              // 19200 bytes
  const int tid = threadIdx.x;
  for (int idx = tid; idx < 150 * 16; idx += 256) {
    int k2 = idx >> 4, n = idx & 15;
    v2f v = {0.f, 0.f};
    if (n < NTAGS) { int k = k2 * 2; v.x = W[n * EMB + k]; v.y = W[n * EMB + k + 1]; }
    wt[idx] = v;
  }
  __syncthreads();

  const int lane  = tid & 31;
  const int wave  = tid >> 5;
  const int tile  = blockIdx.x * 8 + wave;
  const int mrow  = tile * 16 + (lane & 15);
  const int khalf = lane >> 4;              // lanes 16-31 hold K+2,K+3
  const float* rp = emb + (size_t)sentence[mrow] * EMB;

  v8f c = {0.f, 0.f, 0.f, 0.f, 0.f, 0.f, 0.f, 0.f};
  for (int it = 0; it < 75; ++it) {
    // A: 16x4 f32 fragment (row gather, 8B-aligned float2 per lane)
    v2f a = *(const v2f*)(rp + it * 4 + khalf * 2);
    // B: 4x16 f32 fragment from LDS
    v2f b = wt[(it * 2 + khalf) * 16 + (lane & 15)];
    c = __builtin_amdgcn_wmma_f32_16x16x4_f32(false, a, false, b,
                                              (short)0, c, false, false);
  }
  // C/D layout: VGPR r -> M = r + (lane>=16 ? 8 : 0), N = lane&15
  const int n = lane & 15;
  const int mbase = tile * 16 + ((lane >= 16) ? 8 : 0);
#pragma unroll
  for (int r = 0; r < 8; ++r)
    feats[(size_t)(mbase + r) * 16 + n] = c[r];   // cols 14,15 are exact 0
}

// ============================================================
// K_gold: deterministic two-stage gold-score reduction
// ============================================================
__global__ __launch_bounds__(256)
void gold_partial_k(const int* __restrict__ tags,
                    const float* __restrict__ feats,
                    const float* __restrict__ trans,
                    float* __restrict__ partial) {
  __shared__ float red[256];
  float acc = 0.f;
  for (int t = blockIdx.x * 256 + threadIdx.x; t < SEQL; t += 64 * 256) {
    int tg = tags[t];
    int pv = (t == 0) ? STARTT : tags[t - 1];
    acc += trans[tg * NTAGS + pv] + feats[(size_t)t * 16 + tg];
    if (t == SEQL - 1) acc += trans[STOPT * NTAGS + tg];
  }
  red[threadIdx.x] = acc;
  __syncthreads();
  for (int s = 128; s > 0; s >>= 1) {
    if (threadIdx.x < s) red[threadIdx.x] += red[threadIdx.x + s];
    __syncthreads();
  }
  if (threadIdx.x == 0) partial[blockIdx.x] = red[0];
}

// ============================================================
// K2: per-chunk semiring matrix composition (parallel scan, level 1)
//  step matrix A_t[i][j] = trans[i][j] + feat_t[i]
//  Mg = LSE-product over chunk, Pg = max-plus product over chunk
// thread (i,j) of 196 active keeps one matrix element
// ============================================================
__global__ __launch_bounds__(256)
void chunk_scan(const float* __restrict__ feats,
                const float* __restrict__ trans,
                float* __restrict__ Mg, float* __restrict__ Pg) {
  const int g = blockIdx.x, tid = threadIdx.x;
  const int i = tid / NTAGS, j = tid % NTAGS;
  const bool act = (tid < NTAGS * NTAGS);
  __shared__ float Ml[NTAGS * 16];
  __shared__ float Pl[NTAGS * 16];
  __shared__ float fl[16];
  float tr[NTAGS];
  if (act) {
#pragma unroll
    for (int jj = 0; jj < NTAGS; ++jj) tr[jj] = trans[i * NTAGS + jj];
  }
  float m = (act && i == j) ? 0.f : NEGV;   // log-semiring identity
  float p = m;
  const int s0 = g * CHUNK;
  for (int t = s0; t < s0 + CHUNK; ++t) {
    if (act) { Ml[i * 16 + j] = m; Pl[i * 16 + j] = p; }
    if (tid < NTAGS) fl[tid] = feats[(size_t)t * 16 + tid];
    __syncthreads();
    if (act) {
      float mx = -1e30f, px = -1e30f;
#pragma unroll
      for (int jj = 0; jj < NTAGS; ++jj) {
        mx = fmaxf(mx, tr[jj] + Ml[jj * 16 + j]);
        px = fmaxf(px, tr[jj] + Pl[jj * 16 + j]);
      }
      float ss = 0.f;
#pragma unroll
      for (int jj = 0; jj < NTAGS; ++jj)
        ss += __expf(tr[jj] + Ml[jj * 16 + j] - mx);
      m = fl[i] + mx + __logf(ss);
      p = fl[i] + px;
    }
    __syncthreads();
  }
  if (act) { Mg[g * 256 + i * 16 + j] = m; Pg[g * 256 + i * 16 + j] = p; }
}

// ============================================================
// K3: sequential combine across 256 chunks (level 2 of scan),
// single wave, states live in lanes, cross-lane via shuffles.
// Also finalizes forward score, terminal argmax, gold, nll.
// ============================================================
__global__ __launch_bounds__(32)
void combine_k(const float* __restrict__ Mg, const float* __restrict__ Pg,
               const float* __restrict__ trans, const float* __restrict__ gpart,
               float* __restrict__ pvb, int* __restrict__ scal_i,
               float* __restrict__ out) {
  const int lane = threadIdx.x;
  const int i = (lane < NTAGS) ? lane : 0;
  float fv = (lane == STARTT) ? 0.f : NEGV;
  float pv = fv;
  for (int g = 0; g < NCHUNK; ++g) {
    if (lane < 16) pvb[g * 16 + lane] = pv;   // viterbi boundary BEFORE chunk g
    float a[NTAGS], b[NTAGS];
#pragma unroll
    for (int j = 0; j < NTAGS; ++j) {
      float fj = __shfl(fv, j, 32);
      float pj = __shfl(pv, j, 32);
      a[j] = Mg[g * 256 + i * 16 + j] + fj;
      b[j] = Pg[g * 256 + i * 16 + j] + pj;
    }
    float mx = -1e30f, px = -1e30f;
#pragma unroll
    for (int j = 0; j < NTAGS; ++j) { mx = fmaxf(mx, a[j]); px = fmaxf(px, b[j]); }
    float ss = 0.f;
#pragma unroll
    for (int j = 0; j < NTAGS; ++j) ss += __expf(a[j] - mx);
    fv = mx + __logf(ss);
    pv = px;
  }
  // finalize: LSE(fv + trans[STOP]) and argmax(pv + trans[STOP])
  float fvs  = fv + trans[STOPT * NTAGS + i];
  float term = pv + trans[STOPT * NTAGS + i];
  float mx = -1e30f;
#pragma unroll
  for (int j = 0; j < NTAGS; ++j) mx = fmaxf(mx, __shfl(fvs, j, 32));
  float ss = 0.f;
#pragma unroll
  for (int j = 0; j < NTAGS; ++j) ss += __expf(__shfl(fvs, j, 32) - mx);
  float fscore = mx + __logf(ss);
  float bestv = -1e30f; int besti = 0;
#pragma unroll
  for (int j = 0; j < NTAGS; ++j) {
    float tv = __shfl(term, j, 32);
    if (tv > bestv) { bestv = tv; besti = j; }   // first-max wins, like argmax
  }
  if (lane == 0) {
    float gold = 0.f;
    for (int b2 = 0; b2 < 64; ++b2) gold += gpart[b2];   // fixed order: deterministic
    out[0] = fscore - gold;   // nll
    out[1] = bestv;           // path_score
    scal_i[0] = besti;
  }
}

// ============================================================
// K4: per-chunk Viterbi replay from exact boundary vectors.
// Emits packed backpointers (4b/state -> u64 per step) and the
// chunk's composed backward pointer map.
// ============================================================
__global__ __launch_bounds__(32)
void replay_k(const float* __restrict__ feats, const float* __restrict__ trans,
              const float* __restrict__ pvb,
              unsigned long long* __restrict__ bp, int* __restrict__ fmap) {
  const int g = blockIdx.x, lane = threadIdx.x;
  const int n = (lane < NTAGS) ? lane : 0;
  float tr[NTAGS];
#pragma unroll
  for (int p = 0; p < NTAGS; ++p) tr[p] = trans[n * NTAGS + p];
  float pv = (lane < 16) ? pvb[g * 16 + lane] : NEGV;
  __shared__ unsigned long long bpl[CHUNK];
  const int s0 = g * CHUNK;
  for (int t = 0; t < CHUNK; ++t) {
    float pvj[NTAGS];
#pragma unroll
    for (int p = 0; p < NTAGS; ++p) pvj[p] = __shfl(pv, p, 32);
    float best = -1e30f; int arg = 0;
#pragma unroll
    for (int p = 0; p < NTAGS; ++p) {
      float v = pvj[p] + tr[p];
      if (v > best) { best = v; arg = p; }
    }
    unsigned long long pk = 0ull;
#pragma unroll
    for (int p = 0; p < NTAGS; ++p)
      pk |= (unsigned long long)(__shfl(arg, p, 32) & 0xF) << (4 * p);
    if (lane == 0) { bp[s0 + t] = pk; bpl[t] = pk; }
    pv = best + feats[(size_t)(s0 + t) * 16 + n];
  }
  __syncthreads();
  if (lane < NTAGS) {                       // compose backward map of this chunk
    int x = lane;
    for (int t = CHUNK - 1; t >= 0; --t)
      x = (int)((bpl[t] >> (4 * x)) & 0xF);
    fmap[g * 16 + lane] = x;
  }
}

// K5: thread carries across chunk boundaries (256 serial steps)
__global__ void carries_k(const int* __restrict__ fmap, const int* __restrict__ scal_i,
                          int* __restrict__ entry) {
  if (threadIdx.x == 0 && blockIdx.x == 0) {
    int c = scal_i[0];
    for (int g = NCHUNK - 1; g >= 0; --g) {
      entry[g] = c;
      c = fmap[g * 16 + c];
    }
  }
}

// K6: chunk-parallel backtrack emit (path as float, per harness out dtype)
__global__ __launch_bounds__(32)
void backtrack_k(const unsigned long long* __restrict__ bp,
                 const int* __restrict__ entry, float* __restrict__ out_path) {
  const int g = blockIdx.x;
  if (threadIdx.x != 0) return;
  int tag = entry[g];
  const int s0 = g * CHUNK;
  for (int t = CHUNK - 1; t >= 0; --t) {
    out_path[s0 + t] = (float)tag;
    tag = (int)((bp[s0 + t] >> (4 * tag)) & 0xF);
  }
}

extern "C" void kernel_launch(void* const* d_in, const int* in_sizes, int n_in,
                              void* d_out, int out_size, void* d_ws, size_t ws_size,
                              hipStream_t stream) {
  (void)in_sizes; (void)n_in; (void)out_size; (void)ws_size;
  const int*   sentence = (const int*)d_in[0];
  const int*   tags     = (const int*)d_in[1];
  const float* emb      = (const float*)d_in[2];
  const float* W        = (const float*)d_in[3];
  const float* trans    = (const float*)d_in[4];
  float* out = (float*)d_out;

  char* ws = (char*)d_ws;                   // needs ~2.92 MB
  float* feats = (float*)(ws + OFF_FEATS);
  float* Mg    = (float*)(ws + OFF_MG);
  float* Pg    = (float*)(ws + OFF_PG);
  float* pvb   = (float*)(ws + OFF_PVB);
  int*   fmap  = (int*)  (ws + OFF_FMAP);
  int*   entry = (int*)  (ws + OFF_ENTRY);
  unsigned long long* bp = (unsigned long long*)(ws + OFF_BP);
  float* gpart = (float*)(ws + OFF_GOLD);
  int*   scali = (int*)  (ws + OFF_SCALI);

  feats_wmma   <<<SEQL / 16 / 8, 256, 0, stream>>>(sentence, emb, W, feats);
  gold_partial_k<<<64, 256, 0, stream>>>(tags, feats, trans, gpart);
  chunk_scan   <<<NCHUNK, 256, 0, stream>>>(feats, trans, Mg, Pg);
  combine_k    <<<1, 32, 0, stream>>>(Mg, Pg, trans, gpart, pvb, scali, out);
  replay_k     <<<NCHUNK, 32, 0, stream>>>(feats, trans, pvb, bp, fmap);
  carries_k    <<<1, 32, 0, stream>>>(fmap, scali, entry);
  backtrack_k  <<<NCHUNK, 32, 0, stream>>>(bp, entry, out + 2);
}